// SFC_CAE_Encoder_Adaptive_33389075759765
// MI455X (gfx1250) — compile-verified
//
#include <hip/hip_runtime.h>
#include <hip/hip_bf16.h>
#include <math.h>

// ---------------------------------------------------------------------------
// SFC-CAE encoder for gfx1250 (MI455X).
//  - smooth_kernel: gather by SFC ordering + 3-tap smoothing + tanh (mem-bound)
//  - conv_wmma_kernel<CIN>: implicit-im2col GEMM per conv layer using
//    V_WMMA_F32_16X16X4_F32 (fp32 matrix pipe, exact fp32 numerics),
//    dual accumulator chains + global_prefetch of next tile's window
//  - fc1/fc2: tiny VALU dot products
// ---------------------------------------------------------------------------

typedef __attribute__((ext_vector_type(2))) float v2f;
typedef __attribute__((ext_vector_type(8))) float v8f;

static constexpr int NPTS   = 65536;
static constexpr int BATCH  = 32;
static constexpr int NSEQ   = 64;     // SFC(2) * BATCH(32)
static constexpr int KSZ    = 32;
static constexpr int STRIDE = 4;
static constexpr int PADLEN = 16;
static constexpr int WIN    = 92;     // 15*4 + 31 + 1 receptive span for a 16-pos tile
static constexpr int WINP   = 96;     // padded (even, keeps v2f 8B-aligned)

// ---------------------------------------------------------------------------
// Kernel 1: a[s,b,c,n] = tanh( sp_b[n] + sum_k sp_w[n,k] * x[b, ord[s, clip(n+k-1)], c] )
// out layout: [s*32+b][c][n]  (c = COMP = 2)
// ---------------------------------------------------------------------------
__global__ __launch_bounds__(256)
void smooth_kernel(const float* __restrict__ x,      // [B][N][2]
                   const int*   __restrict__ ord,    // [2][N]
                   const float* __restrict__ sp_w,   // [N][3]
                   const float* __restrict__ sp_b,   // [N]
                   float* __restrict__ out)          // [64][2][N]
{
    const int n = blockIdx.x * blockDim.x + threadIdx.x;
    const int s = blockIdx.y;     // 0..1
    const int b = blockIdx.z;     // 0..31
    if (n >= NPTS) return;

    const int nm = (n > 0) ? n - 1 : 0;
    const int np = (n < NPTS - 1) ? n + 1 : NPTS - 1;
    const int j0 = ord[s * NPTS + nm];
    const int j1 = ord[s * NPTS + n];
    const int j2 = ord[s * NPTS + np];

    const float w0 = sp_w[n * 3 + 0];
    const float w1 = sp_w[n * 3 + 1];
    const float w2 = sp_w[n * 3 + 2];
    const float bb = sp_b[n];

    const float2* xb = reinterpret_cast<const float2*>(x) + (size_t)b * NPTS;
    const float2 e0 = xb[j0];
    const float2 e1 = xb[j1];
    const float2 e2 = xb[j2];

    const float a0 = tanhf(w0 * e0.x + w1 * e1.x + w2 * e2.x + bb);
    const float a1 = tanhf(w0 * e0.y + w1 * e1.y + w2 * e2.y + bb);

    const size_t seq = (size_t)(s * BATCH + b);
    out[(seq * 2 + 0) * NPTS + n] = a0;
    out[(seq * 2 + 1) * NPTS + n] = a1;
}

// ---------------------------------------------------------------------------
// Kernel 2: conv1d (K=32, S=4, P=16) + bias + tanh as implicit-im2col GEMM.
//   D[co][pos] = sum_{k=0..CIN*32-1} W[co][k] * X[k][pos]
//   X[k][pos]  = in[ci][ (m0+pos)*4 - 16 + kk ],  k = ci*32 + kk  (zero-padded)
// One wave (=one 32-thread workgroup) computes a 16(co) x 16(pos) tile,
// accumulating over K in steps of 4 via V_WMMA_F32_16X16X4_F32 using two
// independent accumulator chains for matrix-pipe ILP.
//
// LDS:
//   afrag[NG*64] : per-k-group A fragments in exact per-lane layout
//                  afrag[g*64 + l*2 + d] = Wpad[l&15][4g + ((l>>4)<<1) + d]
//   win[CIN*WINP]: zero-padded receptive-field window for the current tile
// ---------------------------------------------------------------------------
template<int CIN>
__global__ __launch_bounds__(32)
void conv_wmma_kernel(const float* __restrict__ in,   // [64][CIN][Lin]
                      const float* __restrict__ w,    // [cout][CIN][32]
                      const float* __restrict__ bias, // [cout]
                      float* __restrict__ out,        // [64][cout][Lout]
                      int Lin, int Lout, int cout)
{
    constexpr int KTOT = CIN * KSZ;
    constexpr int NG   = KTOT / 4;        // number of K=4 steps (even for all layers)

    extern __shared__ float smem[];
    float* afrag = smem;                  // NG*64 floats
    float* win   = smem + NG * 64;        // CIN*WINP floats

    const int seq  = blockIdx.x;          // 0..63
    const int lane = threadIdx.x;         // 0..31

    // ---- build per-lane weight fragments once per block ----
    for (int idx = lane; idx < NG * 32; idx += 32) {
        const int g  = idx >> 5;
        const int l  = idx & 31;
        const int co = l & 15;
        const int kb = 4 * g + ((l >> 4) << 1);
        float v0 = 0.f, v1 = 0.f;
        if (co < cout) {
            const float* wr = w + (size_t)co * KTOT;
            v0 = wr[kb];
            v1 = wr[kb + 1];
        }
        afrag[g * 64 + l * 2 + 0] = v0;
        afrag[g * 64 + l * 2 + 1] = v1;
    }
    __syncthreads();

    const float* inseq = in + (size_t)seq * CIN * Lin;
    const int ntiles   = (Lout + 15) >> 4;

    for (int tile = blockIdx.y; tile < ntiles; tile += gridDim.y) {
        const int m0   = tile << 4;
        const int base = m0 * STRIDE - PADLEN;

        // stage receptive-field window into LDS (zero-padded)
        #pragma unroll
        for (int ci = 0; ci < CIN; ++ci) {
            for (int r = lane; r < WIN; r += 32) {
                const int p = base + r;
                win[ci * WINP + r] = (p >= 0 && p < Lin) ? inseq[(size_t)ci * Lin + p] : 0.f;
            }
        }
        __syncthreads();

        // prefetch next strided tile's window (speculative; overlaps WMMA chain)
        {
            const int tnext = tile + gridDim.y;
            if (tnext < ntiles && lane < 2 * CIN) {
                const int ci   = lane >> 1;
                int p = (tnext << 4) * STRIDE - PADLEN + ((lane & 1) << 6);
                p = (p < 0) ? 0 : ((p >= Lin) ? Lin - 1 : p);
                __builtin_prefetch(&inseq[(size_t)ci * Lin + p], 0, 0);
            }
        }

        // ---- GEMM accumulate: NG fp32 WMMAs over two independent chains ----
        v8f acc0 = {};
        v8f acc1 = {};
        const int pos_l = lane & 15;
        const int sub   = (lane >> 4) << 1;   // 0 for lanes 0-15, 2 for 16-31
        #pragma unroll 8
        for (int g = 0; g < NG; g += 2) {
            {
                const int k0  = g << 2;
                const int ci  = k0 >> 5;
                const int kk0 = k0 & 31;
                const v2f a = *(const v2f*)(afrag + g * 64 + lane * 2);
                const v2f x = *(const v2f*)(win + ci * WINP + pos_l * 4 + kk0 + sub);
                acc0 = __builtin_amdgcn_wmma_f32_16x16x4_f32(
                           false, a, false, x, (short)0, acc0, false, false);
            }
            {
                const int k1  = (g + 1) << 2;
                const int ci  = k1 >> 5;
                const int kk1 = k1 & 31;
                const v2f a = *(const v2f*)(afrag + (g + 1) * 64 + lane * 2);
                const v2f x = *(const v2f*)(win + ci * WINP + pos_l * 4 + kk1 + sub);
                acc1 = __builtin_amdgcn_wmma_f32_16x16x4_f32(
                           false, a, false, x, (short)0, acc1, false, false);
            }
        }
        const v8f acc = acc0 + acc1;

        // ---- bias + tanh + store (lanes 0-15 -> co r, lanes 16-31 -> co r+8) ----
        const int pos    = m0 + pos_l;
        const int cobase = (lane >> 4) << 3;
        if (pos < Lout) {
            #pragma unroll
            for (int r = 0; r < 8; ++r) {
                const int co = cobase + r;
                if (co < cout) {
                    out[((size_t)seq * cout + co) * Lout + pos] = tanhf(acc[r] + bias[co]);
                }
            }
        }
        __syncthreads();   // protect window WAR before next tile
    }
}

// ---------------------------------------------------------------------------
// Kernel 3: h1[b][o] = tanh(fc1_b[o] + sum_i hcat(b,i) * fc1_w[o][i])
//           hcat(b, sfc*1040 + co*65 + pos) = s5[sfc*32+b][co][pos]
// ---------------------------------------------------------------------------
__global__ __launch_bounds__(128)
void fc1_kernel(const float* __restrict__ s5,     // [64][16][65]
                const float* __restrict__ w,      // [128][2080]
                const float* __restrict__ bias,   // [128]
                float* __restrict__ h1)           // [32][128]
{
    const int o = threadIdx.x;   // 0..127
    const int b = blockIdx.x;    // 0..31
    float acc = bias[o];
    const float* wr = w + (size_t)o * 2080;
    int i = 0;
    for (int sfc = 0; sfc < 2; ++sfc) {
        for (int co = 0; co < 16; ++co) {
            const float* src = s5 + (((size_t)(sfc * BATCH + b) * 16 + co) * 65);
            #pragma unroll 5
            for (int pos = 0; pos < 65; ++pos, ++i) {
                acc += src[pos] * wr[i];
            }
        }
    }
    h1[b * 128 + o] = tanhf(acc);
}

__global__ __launch_bounds__(256)
void fc2_kernel(const float* __restrict__ h1,     // [32][128]
                const float* __restrict__ w,      // [16][128]
                const float* __restrict__ bias,   // [16]
                float* __restrict__ out)          // [32][16]
{
    const int t = blockIdx.x * blockDim.x + threadIdx.x;
    if (t >= BATCH * 16) return;
    const int b = t >> 4;
    const int o = t & 15;
    float acc = bias[o];
    const float* wr = w + o * 128;
    const float* hr = h1 + b * 128;
    #pragma unroll 8
    for (int j = 0; j < 128; ++j) acc += hr[j] * wr[j];
    out[t] = tanhf(acc);
}

// ---------------------------------------------------------------------------
// Host launcher
// ---------------------------------------------------------------------------
extern "C" void kernel_launch(void* const* d_in, const int* in_sizes, int n_in,
                              void* d_out, int out_size, void* d_ws, size_t ws_size,
                              hipStream_t stream) {
    const float* x     = (const float*)d_in[0];
    const int*   ord   = (const int*)  d_in[1];
    const float* sp_w  = (const float*)d_in[2];
    const float* sp_b  = (const float*)d_in[3];
    const float* cw[5] = {(const float*)d_in[4], (const float*)d_in[6],
                          (const float*)d_in[8], (const float*)d_in[10],
                          (const float*)d_in[12]};
    const float* cb[5] = {(const float*)d_in[5], (const float*)d_in[7],
                          (const float*)d_in[9], (const float*)d_in[11],
                          (const float*)d_in[13]};
    const float* fc1_w = (const float*)d_in[14];
    const float* fc1_b = (const float*)d_in[15];
    const float* fc2_w = (const float*)d_in[16];
    const float* fc2_b = (const float*)d_in[17];

    // workspace layout (floats), ping-pong across layers (~50.3 MB total)
    float* ws   = (float*)d_ws;
    float* ping = ws;                      // cap 8,388,608 floats (s0 / s2 / s4)
    float* pong = ws + 8388608;            // cap 4,194,560 floats (s1 / s3 / s5)
    float* h1   = pong + 4194560;          // 32*128 floats

    // 1) gather + smooth + tanh -> s0 = ping : [64][2][65536]
    smooth_kernel<<<dim3((NPTS + 255) / 256, 2, BATCH), 256, 0, stream>>>(
        x, ord, sp_w, sp_b, ping);

    // 2) conv stack (each: dyn-LDS = (NG*64 + CIN*WINP) * 4 bytes)
    // L0: [64][2][65536] -> [64][4][16385], tiles=1025
    conv_wmma_kernel<2><<<dim3(NSEQ, 128), 32, ( 16 * 64 +  2 * WINP) * 4, stream>>>(
        ping, cw[0], cb[0], pong, 65536, 16385, 4);
    // L1: [64][4][16385] -> [64][8][4097], tiles=257
    conv_wmma_kernel<4><<<dim3(NSEQ, 128), 32, ( 32 * 64 +  4 * WINP) * 4, stream>>>(
        pong, cw[1], cb[1], ping, 16385, 4097, 8);
    // L2: [64][8][4097] -> [64][16][1025], tiles=65
    conv_wmma_kernel<8><<<dim3(NSEQ, 65), 32, ( 64 * 64 +  8 * WINP) * 4, stream>>>(
        ping, cw[2], cb[2], pong, 4097, 1025, 16);
    // L3: [64][16][1025] -> [64][16][257], tiles=17
    conv_wmma_kernel<16><<<dim3(NSEQ, 17), 32, (128 * 64 + 16 * WINP) * 4, stream>>>(
        pong, cw[3], cb[3], ping, 1025, 257, 16);
    // L4: [64][16][257] -> [64][16][65], tiles=5
    conv_wmma_kernel<16><<<dim3(NSEQ, 5), 32, (128 * 64 + 16 * WINP) * 4, stream>>>(
        ping, cw[4], cb[4], pong, 257, 65, 16);

    // 3) FC head
    fc1_kernel<<<BATCH, 128, 0, stream>>>(pong, fc1_w, fc1_b, h1);
    fc2_kernel<<<2, 256, 0, stream>>>(h1, fc2_w, fc2_b, (float*)d_out);
}